// DynamicRouting_74517682586059
// MI455X (gfx1250) — compile-verified
//
#include <hip/hip_runtime.h>
#include <math.h>

typedef __attribute__((ext_vector_type(2))) float v2f;
typedef __attribute__((ext_vector_type(4))) float v4f;
typedef __attribute__((ext_vector_type(8))) float v8f;

#define NG    8
#define NFO   64
#define NFI   64
#define NB    32
#define NHW   4096
#define TILE  16           // spatial positions per WMMA tile (M dim)
#define TILES 8            // tiles per workgroup
#define POSWG (TILE*TILES) // 128 positions per workgroup

__global__ __launch_bounds__(256) void dynroute_wmma(
    const float* __restrict__ x,     // [B, G*FI, H*W]
    const float* __restrict__ w,     // [G, FO, FI]
    const float* __restrict__ bias,  // [FO]
    float* __restrict__ out)         // [B, FO, H*W]
{
  __shared__ float con[TILE][NG][NFO];  // 32 KB: con for current tile
  __shared__ float vout[NFO][TILE];     // 4 KB: output staging (transpose)

  const int tid  = threadIdx.x;
  const int g    = tid >> 5;   // wave id == group id
  const int lane = tid & 31;
  const int li   = lane & 15;  // low-lane index (M for A, N for B)
  const int hi   = lane >> 4;  // lane-half (selects K pair)

  const int b       = blockIdx.x >> 5;           // 32 WGs per image
  const int posBase = (blockIdx.x & 31) * POSWG;

  // ---- preload B fragments: weight[g] (16 KB) into 128 VGPRs ----
  // frag(k,n): this lane holds B[K = 4k + 2*hi + {0,1}][N = 16n + li]
  // weight[g][o][i] with i consecutive -> single b64 per fragment.
  v2f bf[16][4];
  {
    const float* wg = w + g * (NFO * NFI);
    #pragma unroll
    for (int k = 0; k < 16; ++k)
      #pragma unroll
      for (int n = 0; n < 4; ++n)
        bf[k][n] = *(const v2f*)(wg + (n * 16 + li) * NFI + k * 4 + hi * 2);
  }

  const float* xg   = x + ((size_t)b * (NG * NFI) + (size_t)g * NFI) * NHW + posBase;
  float*       outb = out + (size_t)b * NFO * NHW + posBase;

  // routing-phase mapping: p = position in tile, fo0 = 4-wide fo slice
  const int p   = tid >> 4;
  const int fo0 = (tid & 15) * 4;
  const v4f bi  = *(const v4f*)(bias + fo0);

  for (int t = 0; t < TILES; ++t) {
    const int tpos = t * TILE;

    // ================= compute phase: con = x_tile @ w[g]^T =================
    v8f acc[4] = {};
    #pragma unroll
    for (int k = 0; k < 16; ++k) {
      // A frag: lane holds A[M = li][K = 4k + 2*hi + {0,1}]
      const float* ap = xg + (size_t)(k * 4 + hi * 2) * NHW + tpos + li;
      v2f a;
      a.x = ap[0];
      a.y = ap[NHW];
      if (t + 1 < TILES) {  // uniform branch: prefetch next tile's lines
        __builtin_prefetch(ap + TILE, 0, 3);
        __builtin_prefetch(ap + NHW + TILE, 0, 3);
      }
      #pragma unroll
      for (int n = 0; n < 4; ++n) {
        acc[n] = __builtin_amdgcn_wmma_f32_16x16x4_f32(
            false, a, false, bf[k][n], (short)0, acc[n], false, false);
      }
    }

    // C layout: VGPR r -> M = r (lanes 0-15) / r+8 (lanes 16-31), N = li
    #pragma unroll
    for (int n = 0; n < 4; ++n)
      #pragma unroll
      for (int r = 0; r < 8; ++r)
        con[r + hi * 8][g][n * 16 + li] = acc[n][r];

    __syncthreads();

    // ================= routing phase (3 iterations, on-chip) ================
    v4f cg[8];
    #pragma unroll
    for (int gg = 0; gg < NG; ++gg)
      cg[gg] = *(const v4f*)&con[p][gg][fo0];

    float beta[NG] = {};
    v4f v = {};
    #pragma unroll
    for (int it = 0; it < 3; ++it) {
      v4f vn = {};
      #pragma unroll
      for (int gg = 0; gg < NG; ++gg) {
        const float alpha = 1.0f / (1.0f + __expf(-beta[gg]));
        vn += cg[gg] * alpha;
      }
      v = vn;
      if (it == 2) break;
      #pragma unroll
      for (int gg = 0; gg < NG; ++gg) {
        float s = v.x * cg[gg].x + v.y * cg[gg].y +
                  v.z * cg[gg].z + v.w * cg[gg].w;
        // reduce across the 16 lanes that share this position
        s += __shfl_xor(s, 1, 32);
        s += __shfl_xor(s, 2, 32);
        s += __shfl_xor(s, 4, 32);
        s += __shfl_xor(s, 8, 32);
        beta[gg] += s;
      }
    }

    // stage v + bias into LDS transposed so the global store coalesces
    vout[fo0 + 0][p] = v.x + bi.x;
    vout[fo0 + 1][p] = v.y + bi.y;
    vout[fo0 + 2][p] = v.z + bi.z;
    vout[fo0 + 3][p] = v.w + bi.w;
    __syncthreads();

    // ================= coalesced b128 store ================
    {
      const int fo = tid >> 2;
      const int p4 = (tid & 3) * 4;
      v4f r = *(const v4f*)&vout[fo][p4];
      *(v4f*)(outb + (size_t)fo * NHW + tpos + p4) = r;
    }
    __syncthreads();  // protect con/vout before next tile overwrites
  }
}

extern "C" void kernel_launch(void* const* d_in, const int* in_sizes, int n_in,
                              void* d_out, int out_size, void* d_ws, size_t ws_size,
                              hipStream_t stream) {
  const float* x    = (const float*)d_in[0];
  const float* wgt  = (const float*)d_in[1];
  const float* bias = (const float*)d_in[2];
  float* out        = (float*)d_out;
  (void)in_sizes; (void)n_in; (void)out_size; (void)d_ws; (void)ws_size;

  dim3 grid(NB * (NHW / POSWG));  // 32 * 32 = 1024 workgroups
  dim3 block(256);                // 8 waves = 8 groups
  dynroute_wmma<<<grid, block, 0, stream>>>(x, wgt, bias, out);
}